// Net_51642686767930
// MI455X (gfx1250) — compile-verified
//
#include <hip/hip_runtime.h>
#include <hip/hip_bf16.h>

// ---------------------------------------------------------------------------
// MI455X (gfx1250, wave32) implementation.
//
// Part 1 (STCConv): atomic scatter of neighbor features (~0.8 GB through
//   L2 @ 23.3 TB/s -> bandwidth bound), then one fused WMMA-f32 GEMM
//   (100000x128 @ 128x128) + bias + ReLU + row log-softmax in LDS.
// Part 2 (RGCN): re-associate the per-basis scatter into coefficient
//   matrices A_b[d,s] = sum coeff (6M atomics, 126 MB buffer, L2-resident),
//   then pure WMMA-f32 GEMMs with LDS-staged B tiles (async global->LDS
//   copies via GLOBAL_LOAD_ASYNC_TO_LDS_B128 + s_wait_asynccnt):
//      M_b = x_g2 @ basis_b   (grid.z = 31 also computes H = x_g2 @ root)
//      P  += A_b @ M_b        (grid.z = 30, fp32 atomic epilogue into P)
//   epilogue: relu(H + bias + P/deg) -> row log-softmax (1024-wide).
// ---------------------------------------------------------------------------

typedef float v2f __attribute__((ext_vector_type(2)));
typedef float v8f __attribute__((ext_vector_type(8)));
typedef int   v4i __attribute__((vector_size(16)));   // matches builtin param

#define N1C 100000
#define E1C 1600000
#define DC  128
#define N2C 1024
#define E2C 200000
#define NBC 30

#define KT  32      // K-tile staged in LDS
#define BSP 132     // padded LDS row (128 + 4)

#if defined(__gfx1250__) && __has_builtin(__builtin_amdgcn_global_load_async_to_lds_b128)
#define USE_ASYNC_LDS 1
#else
#define USE_ASYNC_LDS 0
#endif

__device__ __forceinline__ void wait_async_lds() {
#if USE_ASYNC_LDS
#if __has_builtin(__builtin_amdgcn_s_wait_asynccnt)
  __builtin_amdgcn_s_wait_asynccnt(0);
#else
  asm volatile("s_wait_asynccnt 0x0" ::: "memory");
#endif
#endif
}

// ---------------- generic utility kernels ----------------------------------

__global__ void zero_f32(float* __restrict__ p, long n) {
  long stride = (long)gridDim.x * blockDim.x;
  for (long i = (long)blockIdx.x * blockDim.x + threadIdx.x; i < n; i += stride)
    p[i] = 0.0f;
}

__global__ void deg_count(const int* __restrict__ dst, float* __restrict__ deg, int E) {
  int stride = gridDim.x * blockDim.x;
  for (int e = blockIdx.x * blockDim.x + threadIdx.x; e < E; e += stride)
    atomicAdd(deg + dst[e], 1.0f);
}

// ---------------- part 1: feature scatter -----------------------------------
// thread t handles (edge = t/32, 4-float chunk = t%32): 128 floats per edge.
__global__ void scatter_agg(const int* __restrict__ src, const int* __restrict__ dst,
                            const float* __restrict__ x, float* __restrict__ agg,
                            long total) {
  long stride = (long)gridDim.x * blockDim.x;
  for (long t = (long)blockIdx.x * blockDim.x + threadIdx.x; t < total; t += stride) {
    int e = (int)(t >> 5);
    int c = (int)(t & 31);
    int s = src[e], d = dst[e];
    const float4 xv = ((const float4*)(x + (size_t)s * DC))[c];
    float* ap = agg + (size_t)d * DC + c * 4;
    atomicAdd(ap + 0, xv.x);
    atomicAdd(ap + 1, xv.y);
    atomicAdd(ap + 2, xv.z);
    atomicAdd(ap + 3, xv.w);
  }
}

// ---------------- part 1: fused GEMM + bias + relu + log-softmax ------------
// block = 256 threads = 8 wave32; block covers 16 rows x 128 cols.
// wave w computes the 16x16 tile at columns [16w, 16w+16) via WMMA f32 16x16x4.
__global__ __launch_bounds__(256) void stc_gemm_softmax(
    const float* __restrict__ agg, const float* __restrict__ deg,
    const float* __restrict__ W, const float* __restrict__ bias,
    float* __restrict__ out) {
  __shared__ float hsm[16][DC + 4];
  __shared__ float red[16][17];

  const int tid  = threadIdx.x;
  const int wave = tid >> 5;
  const int lane = tid & 31;
  const int half = lane >> 4;   // 0: K+0/K+1 ; 1: K+2/K+3
  const int idx  = lane & 15;   // A row (M) / B col (N)
  const int row0 = blockIdx.x * 16;
  const int n0   = wave * 16;

  const float* arow = agg + (size_t)(row0 + idx) * DC;
  const float invd = 1.0f / fmaxf(deg[row0 + idx], 1.0f);   // mean-aggregate

  v8f c = {};
  for (int k = 0; k < DC; k += 4) {
    v2f a, b;
    a.x = arow[k + 2 * half] * invd;
    a.y = arow[k + 2 * half + 1] * invd;
    b.x = W[(size_t)(k + 2 * half) * DC + n0 + idx];
    b.y = W[(size_t)(k + 2 * half + 1) * DC + n0 + idx];
    c = __builtin_amdgcn_wmma_f32_16x16x4_f32(false, a, false, b, (short)0, c,
                                              false, false);
  }

  const float bn = bias[n0 + idx];
#pragma unroll
  for (int j = 0; j < 8; ++j)
    hsm[j + 8 * half][n0 + idx] = fmaxf(c[j] + bn, 0.0f);
  __syncthreads();

  // log-softmax: 16 threads per row, 8 columns each
  const int r = tid >> 4;
  const int t = tid & 15;
  float mx = -3.402823466e38f;
#pragma unroll
  for (int q = 0; q < 8; ++q) mx = fmaxf(mx, hsm[r][t * 8 + q]);
  red[r][t] = mx;
  __syncthreads();
  if (t == 0) {
    float m2 = red[r][0];
#pragma unroll
    for (int q = 1; q < 16; ++q) m2 = fmaxf(m2, red[r][q]);
    red[r][16] = m2;
  }
  __syncthreads();
  const float rowmax = red[r][16];
  float se = 0.0f;
#pragma unroll
  for (int q = 0; q < 8; ++q) se += __expf(hsm[r][t * 8 + q] - rowmax);
  __syncthreads();
  red[r][t] = se;
  __syncthreads();
  if (t == 0) {
    float s2 = 0.0f;
#pragma unroll
    for (int q = 0; q < 16; ++q) s2 += red[r][q];
    red[r][16] = __logf(s2);
  }
  __syncthreads();
  const float lse = red[r][16];
  float* orow = out + (size_t)(row0 + r) * DC;
#pragma unroll
  for (int q = 0; q < 8; ++q)
    orow[t * 8 + q] = hsm[r][t * 8 + q] - rowmax - lse;
}

// ---------------- part 2: coefficient scatter into A_b ----------------------
__global__ void scatter_coeff(const int* __restrict__ src, const int* __restrict__ dst,
                              const int* __restrict__ etype, const float* __restrict__ att,
                              float* __restrict__ Abuf, int E) {
  int stride = gridDim.x * blockDim.x;
  for (int e = blockIdx.x * blockDim.x + threadIdx.x; e < E; e += stride) {
    int s = src[e], d = dst[e], r = etype[e];
    const float* co = att + (size_t)r * NBC;
    float* base = Abuf + (size_t)d * N2C + s;
#pragma unroll 5
    for (int b = 0; b < NBC; ++b)
      atomicAdd(base + (size_t)b * N2C * N2C, co[b]);
  }
}

// ---------------- LDS-staged 128x128 WMMA GEMM building blocks --------------

// Stage B[k0:k0+KT, 0:128] (column-offset folded into B) into Bs.
// 256 threads move 32x128 floats as 1024 x b128 transfers (4 per thread).
__device__ __forceinline__ void stage_B_tile(const float* __restrict__ B, int ldb,
                                             int k0, float (*Bs)[BSP], int tid) {
#pragma unroll
  for (int i = 0; i < 4; ++i) {
    int lin = tid + i * 256;      // 0..1023
    int r   = lin >> 5;           // 0..31
    int c   = (lin & 31) * 4;     // 0..124
    const float* g = B + (size_t)(k0 + r) * ldb + c;
#if USE_ASYNC_LDS
    __builtin_amdgcn_global_load_async_to_lds_b128(
        (__attribute__((address_space(1))) v4i*)(void*)g,
        (__attribute__((address_space(3))) v4i*)(void*)&Bs[r][c], 0, 0);
#else
    const float4 tv = *(const float4*)g;
    *(float4*)&Bs[r][c] = tv;
#endif
  }
}

// One K-tile of compute: wave's 16 rows x 128 cols (8 accumulators).
__device__ __forceinline__ void compute_tile(const float* __restrict__ arow,
                                             float (*Bs)[BSP], int half, int idx,
                                             v8f c[8]) {
#pragma unroll
  for (int ks = 0; ks < KT; ks += 4) {
    v2f a;
    a.x = arow[ks + 2 * half];
    a.y = arow[ks + 2 * half + 1];
#pragma unroll
    for (int sub = 0; sub < 8; ++sub) {
      v2f b;
      b.x = Bs[ks + 2 * half][sub * 16 + idx];
      b.y = Bs[ks + 2 * half + 1][sub * 16 + idx];
      c[sub] = __builtin_amdgcn_wmma_f32_16x16x4_f32(false, a, false, b, (short)0,
                                                     c[sub], false, false);
    }
  }
}

// ---------------- part 2: batched M_b = X @ basis_b  (z==NB -> H = X @ root)
// grid (8, 8, 31); block = 8 waves; block tile 128x128; B tile staged in LDS.
__global__ __launch_bounds__(256) void g2_gemm_xb(
    const float* __restrict__ X, const float* __restrict__ basis,
    const float* __restrict__ root, float* __restrict__ Mbuf,
    float* __restrict__ Hbuf) {
  __shared__ float Bs[KT][BSP];
  const int z = blockIdx.z;
  const float* B = (z < NBC) ? basis + (size_t)z * N2C * N2C : root;
  float*       O = (z < NBC) ? Mbuf  + (size_t)z * N2C * N2C : Hbuf;

  const int tid  = threadIdx.x;
  const int wave = tid >> 5;
  const int lane = tid & 31;
  const int half = lane >> 4;
  const int idx  = lane & 15;
  const int row0 = blockIdx.x * 128;
  const int col0 = blockIdx.y * 128;

  const float* arow = X + (size_t)(row0 + wave * 16 + idx) * N2C;
  v8f c[8] = {};
  for (int k0 = 0; k0 < N2C; k0 += KT) {
    stage_B_tile(B + col0, N2C, k0, Bs, tid);
    wait_async_lds();
    __syncthreads();
    compute_tile(arow + k0, Bs, half, idx, c);
    __syncthreads();
  }
#pragma unroll
  for (int sub = 0; sub < 8; ++sub)
#pragma unroll
    for (int j = 0; j < 8; ++j)
      O[(size_t)(row0 + wave * 16 + j + 8 * half) * N2C + col0 + sub * 16 + idx] =
          c[sub][j];
}

// ---------------- part 2: P += A_b @ M_b per z-slice (K = 1024 each) --------
// grid (8, 8, 30); fp32 atomic accumulation into the 4 MB L2-resident P.
__global__ __launch_bounds__(256) void g2_gemm_am(
    const float* __restrict__ Abuf, const float* __restrict__ Mbuf,
    float* __restrict__ P) {
  __shared__ float Bs[KT][BSP];
  const int z = blockIdx.z;
  const float* A = Abuf + (size_t)z * N2C * N2C;
  const float* B = Mbuf + (size_t)z * N2C * N2C;

  const int tid  = threadIdx.x;
  const int wave = tid >> 5;
  const int lane = tid & 31;
  const int half = lane >> 4;
  const int idx  = lane & 15;
  const int row0 = blockIdx.x * 128;
  const int col0 = blockIdx.y * 128;

  const float* arow = A + (size_t)(row0 + wave * 16 + idx) * N2C;
  v8f c[8] = {};
  for (int k0 = 0; k0 < N2C; k0 += KT) {
    stage_B_tile(B + col0, N2C, k0, Bs, tid);
    wait_async_lds();
    __syncthreads();
    compute_tile(arow + k0, Bs, half, idx, c);
    __syncthreads();
  }
#pragma unroll
  for (int sub = 0; sub < 8; ++sub)
#pragma unroll
    for (int j = 0; j < 8; ++j)
      atomicAdd(&P[(size_t)(row0 + wave * 16 + j + 8 * half) * N2C + col0 +
                   sub * 16 + idx],
                c[sub][j]);
}

// ---------------- part 2: epilogue relu + log-softmax (1024-wide rows) ------
__global__ __launch_bounds__(256) void rgcn_epilogue(
    const float* __restrict__ H, const float* __restrict__ P,
    const float* __restrict__ deg2, const float* __restrict__ bias,
    float* __restrict__ out) {
  __shared__ float red[256];
  const int row = blockIdx.x;
  const int tid = threadIdx.x;
  const float invd = 1.0f / fmaxf(deg2[row], 1.0f);

  float v[4];
  float mx = -3.402823466e38f;
#pragma unroll
  for (int q = 0; q < 4; ++q) {
    int n = tid + q * 256;
    size_t o = (size_t)row * N2C + n;
    float h = fmaxf(H[o] + bias[n] + P[o] * invd, 0.0f);
    v[q] = h;
    mx = fmaxf(mx, h);
  }
  red[tid] = mx;
  __syncthreads();
  for (int s = 128; s > 0; s >>= 1) {
    if (tid < s) red[tid] = fmaxf(red[tid], red[tid + s]);
    __syncthreads();
  }
  const float rowmax = red[0];
  __syncthreads();
  float se = 0.0f;
#pragma unroll
  for (int q = 0; q < 4; ++q) se += __expf(v[q] - rowmax);
  red[tid] = se;
  __syncthreads();
  for (int s = 128; s > 0; s >>= 1) {
    if (tid < s) red[tid] += red[tid + s];
    __syncthreads();
  }
  const float lse = __logf(red[0]);
#pragma unroll
  for (int q = 0; q < 4; ++q)
    out[(size_t)row * N2C + tid + q * 256] = v[q] - rowmax - lse;
}

// ---------------------------------------------------------------------------

extern "C" void kernel_launch(void* const* d_in, const int* in_sizes, int n_in,
                              void* d_out, int out_size, void* d_ws, size_t ws_size,
                              hipStream_t stream) {
  const float* x_g1   = (const float*)d_in[0];
  const float* W_stc  = (const float*)d_in[1];
  const float* b_stc  = (const float*)d_in[2];
  const float* x_g2   = (const float*)d_in[3];
  const float* basis  = (const float*)d_in[4];
  const float* att    = (const float*)d_in[5];
  const float* root   = (const float*)d_in[6];
  const float* b_rg   = (const float*)d_in[7];
  const int*   ei1    = (const int*)d_in[8];   // [2, E1]: src row then dst row
  const int*   ei2    = (const int*)d_in[9];   // [2, E2]
  const int*   etype  = (const int*)d_in[10];  // [E2]

  float* out1 = (float*)d_out;                       // [N1, 128]
  float* out2 = (float*)d_out + (size_t)N1C * DC;    // [N2, N2]

  // workspace layout (floats, 64-float aligned)
  float* ws = (float*)d_ws;
  const size_t deg1_f = 0;                               // 100000
  const size_t deg2_f = deg1_f + 100032;                 // 1024
  const size_t agg1_f = deg2_f + 1024;                   // 12,800,000
  const size_t A_f    = agg1_f + (size_t)N1C * DC;       // 30*1024*1024
  const size_t M_f    = A_f + (size_t)NBC * N2C * N2C;   // 30*1024*1024
  const size_t P_f    = M_f + (size_t)NBC * N2C * N2C;   // 1024*1024
  const size_t H_f    = P_f + (size_t)N2C * N2C;         // 1024*1024
  float* deg1 = ws + deg1_f;
  float* deg2 = ws + deg2_f;
  float* agg1 = ws + agg1_f;
  float* Abuf = ws + A_f;
  float* Mbuf = ws + M_f;
  float* Pbuf = ws + P_f;
  float* Hbuf = ws + H_f;

  // zero the atomically-accumulated regions: deg1, deg2, agg1, Abuf (+ Pbuf)
  const long zero_n = (long)(A_f + (size_t)NBC * N2C * N2C);
  zero_f32<<<dim3(16384), dim3(256), 0, stream>>>(ws, zero_n);
  zero_f32<<<dim3(1024), dim3(256), 0, stream>>>(Pbuf, (long)N2C * N2C);

  // ---- part 1 ----
  deg_count<<<dim3(4096), dim3(256), 0, stream>>>(ei1 + E1C, deg1, E1C);
  scatter_agg<<<dim3(65536), dim3(256), 0, stream>>>(ei1, ei1 + E1C, x_g1, agg1,
                                                     (long)E1C * 32);
  stc_gemm_softmax<<<dim3(N1C / 16), dim3(256), 0, stream>>>(agg1, deg1, W_stc,
                                                             b_stc, out1);

  // ---- part 2 ----
  deg_count<<<dim3(1024), dim3(256), 0, stream>>>(ei2 + E2C, deg2, E2C);
  scatter_coeff<<<dim3(1024), dim3(256), 0, stream>>>(ei2, ei2 + E2C, etype, att,
                                                      Abuf, E2C);
  g2_gemm_xb<<<dim3(N2C / 128, N2C / 128, NBC + 1), dim3(256), 0, stream>>>(
      x_g2, basis, root, Mbuf, Hbuf);
  g2_gemm_am<<<dim3(N2C / 128, N2C / 128, NBC), dim3(256), 0, stream>>>(Abuf, Mbuf,
                                                                        Pbuf);
  rgcn_epilogue<<<dim3(N2C), dim3(256), 0, stream>>>(Hbuf, Pbuf, deg2, b_rg, out2);
}